// gradientfree_4535485464998
// MI455X (gfx1250) — compile-verified
//
#include <hip/hip_runtime.h>

// Problem constants (match reference).
#define N_PTS 16384   // 128*128
#define NF    128
#define NR    9
#define BATCH 64
#define NU_C    0.08f
#define ALPHA_C 0.5f

typedef __attribute__((ext_vector_type(2))) float v2f;
typedef __attribute__((ext_vector_type(8))) float v8f;

// Full-block (256-thread, 8 waves) sum reduction.
// Wave-level stage uses V_WMMA_F32_16X16X4_F32:
//   A(16x4): VGPR0 lanes0-15 = K=0 rows, lanes16-31 = K=2 rows; VGPR1 = K1/K3 (zero).
//   With a = {v, 0} and B = all-ones, D[m,n] = v[m] + v[m+16] (replicated over n).
//   D layout: vgpr j, lane i<16 -> D[j, i]; lane 16+i -> D[8+j, i].
//   So lane0 sums D[0..7,0], lane16 sums D[8..15,0]; their sum = wave total.
// Requires EXEC all-ones: callers only invoke with full waves, no prior divergence.
__device__ __forceinline__ float block_reduce_256(float v) {
    __shared__ float red[16];
    v2f a; a.x = v;    a.y = 0.0f;
    v2f b; b.x = 1.0f; b.y = 1.0f;
    v8f c = {};
    c = __builtin_amdgcn_wmma_f32_16x16x4_f32(
            /*neg_a=*/false, a, /*neg_b=*/false, b,
            /*c_mod=*/(short)0, c, /*reuse_a=*/false, /*reuse_b=*/false);
    float s = c[0] + c[1] + c[2] + c[3] + c[4] + c[5] + c[6] + c[7];
    const int lane = threadIdx.x & 31;
    const int wave = threadIdx.x >> 5;
    if (lane == 0)  red[wave * 2 + 0] = s;
    if (lane == 16) red[wave * 2 + 1] = s;
    __syncthreads();
    if (threadIdx.x == 0) {
        float total = 0.0f;
#pragma unroll
        for (int i = 0; i < 16; ++i) total += red[i];
        red[0] = total;
    }
    __syncthreads();
    float r = red[0];
    __syncthreads();   // allow safe reuse of `red` on a second call
    return r;
}

// Kernel 0: fold xd (neighbor coord diffs) with inv_mat into per-tap weights.
// w0[r*N+n] = dx*inv[0][0] + dy*inv[1][0]   (x-derivative weight)
// w1[r*N+n] = dx*inv[0][1] + dy*inv[1][1]   (y-derivative weight)
// [r][n] layout -> fully coalesced loads in the stencil passes.
__global__ void k_weights(const float* __restrict__ xx,
                          const float* __restrict__ inv_mat,
                          const int*   __restrict__ n_index,
                          float* __restrict__ w0,
                          float* __restrict__ w1) {
    const int n = blockIdx.x * 256 + threadIdx.x;   // grid covers exactly N_PTS
    const float xc  = xx[2 * n];
    const float yc  = xx[2 * n + 1];
    const float i00 = inv_mat[4 * n + 0];
    const float i01 = inv_mat[4 * n + 1];
    const float i10 = inv_mat[4 * n + 2];
    const float i11 = inv_mat[4 * n + 3];
#pragma unroll
    for (int r = 0; r < NR; ++r) {
        const int nb = n_index[n * NR + r];
        const float dx = xx[2 * nb]     - xc;
        const float dy = xx[2 * nb + 1] - yc;
        w0[r * N_PTS + n] = dx * i00 + dy * i10;
        w1[r * N_PTS + n] = dx * i01 + dy * i11;
    }
}

// Kernel 1: first-derivative pass, u_x[b,n] = sum_r (u[b,nbr]-u[b,n]) * w0[n,r].
__global__ void k_ux(const float* __restrict__ yp,
                     const int*   __restrict__ n_index,
                     const float* __restrict__ w0,
                     float* __restrict__ ux) {
    const int idx = blockIdx.x * 256 + threadIdx.x;       // 0 .. B*N-1
    const int n   = idx & (N_PTS - 1);
    const float* __restrict__ u = yp + (size_t)(idx - n); // batch base (b<<14)
    const float uc = u[n];
    float acc = 0.0f;
#pragma unroll
    for (int r = 0; r < NR; ++r) {
        const int nb = n_index[n * NR + r];
        acc += (u[nb] - uc) * w0[r * N_PTS + n];
    }
    ux[idx] = acc;
}

// Kernel 2: residual pass. u_y from u with w1; u_xx from u_x with w0;
// f = u_y - NU*u_xx - u(1-u)(u+alpha); block-reduce f^2 -> one partial per block.
__global__ void k_resid(const float* __restrict__ yp,
                        const int*   __restrict__ n_index,
                        const float* __restrict__ w0,
                        const float* __restrict__ w1,
                        const float* __restrict__ ux,
                        float* __restrict__ partials) {
    const int idx = blockIdx.x * 256 + threadIdx.x;
    const int n   = idx & (N_PTS - 1);
    const size_t base = (size_t)(idx - n);
    const float* __restrict__ u   = yp + base;
    const float* __restrict__ uxb = ux + base;
    const float uc  = u[n];
    const float uxc = uxb[n];
    float uy = 0.0f, uxx = 0.0f;
#pragma unroll
    for (int r = 0; r < NR; ++r) {
        const int nb = n_index[n * NR + r];
        uy  += (u[nb]   - uc)  * w1[r * N_PTS + n];
        uxx += (uxb[nb] - uxc) * w0[r * N_PTS + n];
    }
    const float f = uy - NU_C * uxx - uc * (1.0f - uc) * (uc + ALPHA_C);
    const float t = block_reduce_256(f * f);
    if (threadIdx.x == 0) partials[blockIdx.x] = t;
}

// Kernel 3: boundary MSE partials. Elements: top row, left col, right col (384/batch).
__global__ void k_bnd(const float* __restrict__ yp,
                      const float* __restrict__ yd,
                      float* __restrict__ partials) {
    const int idx = blockIdx.x * 256 + threadIdx.x;   // 0 .. B*384-1 (= 96 blocks)
    const int b = idx / 384;
    const int m = idx - b * 384;
    int n;
    if (m < 128)      n = m;                          // row i=0
    else if (m < 256) n = (m - 128) * NF;             // col j=0
    else              n = (m - 256) * NF + (NF - 1);  // col j=127
    const size_t off = ((size_t)b << 14) + (size_t)n;
    const float d = yp[off] - yd[off];
    const float t = block_reduce_256(d * d);
    if (threadIdx.x == 0) partials[blockIdx.x] = t;
}

// Kernel 4: deterministic final sum of both partial arrays -> scalar loss.
__global__ void k_final(const float* __restrict__ pf,   // 4096 residual partials
                        const float* __restrict__ pu,   // 96 boundary partials
                        float* __restrict__ out) {
    float sf = 0.0f;
    for (int i = threadIdx.x; i < 4096; i += 256) sf += pf[i];
    sf = block_reduce_256(sf);
    float su = (threadIdx.x < 96) ? pu[threadIdx.x] : 0.0f;
    su = block_reduce_256(su);
    if (threadIdx.x == 0) {
        out[0] = sf / (float)(BATCH * N_PTS) + su / (float)(BATCH * 384);
    }
}

extern "C" void kernel_launch(void* const* d_in, const int* in_sizes, int n_in,
                              void* d_out, int out_size, void* d_ws, size_t ws_size,
                              hipStream_t stream) {
    (void)in_sizes; (void)n_in; (void)out_size; (void)ws_size;
    const float* yp = (const float*)d_in[0];   // y_pred  [B,128,128]
    const float* yd = (const float*)d_in[1];   // y_dash  [B,128,128]
    const float* xx = (const float*)d_in[2];   // xx      [N,2]
    const float* im = (const float*)d_in[3];   // inv_mat [N,2,2]
    const int*   ni = (const int*)  d_in[4];   // n_index [N,9]
    float* out = (float*)d_out;

    // Workspace layout (floats): w0 | w1 | ux | partials_f | partials_u  (~5.4 MB)
    float* ws = (float*)d_ws;
    float* w0 = ws;
    float* w1 = w0 + (size_t)N_PTS * NR;
    float* ux = w1 + (size_t)N_PTS * NR;
    float* pf = ux + (size_t)BATCH * N_PTS;
    float* pu = pf + 4096;

    k_weights<<<N_PTS / 256,            256, 0, stream>>>(xx, im, ni, w0, w1);
    k_ux     <<<(BATCH * N_PTS) / 256,  256, 0, stream>>>(yp, ni, w0, ux);
    k_resid  <<<(BATCH * N_PTS) / 256,  256, 0, stream>>>(yp, ni, w0, w1, ux, pf);
    k_bnd    <<<(BATCH * 384) / 256,    256, 0, stream>>>(yp, yd, pu);
    k_final  <<<1,                      256, 0, stream>>>(pf, pu, out);
}